// LinearTransmissionLayer_32298154066214
// MI455X (gfx1250) — compile-verified
//
#include <hip/hip_runtime.h>

// ---------------------------------------------------------------------------
// mean(X, axis=-2) @ B + cT, broadcast to all N rows.
//   X  : [16, 4096, 1024] f32
//   B  : [1024, 1024]     f32
//   cT : [1, 1024]        f32
//   out: [16, 4096, 1024] f32
// Bandwidth-bound: 256MB read + 256MB write ~ 22us @ 23.3 TB/s.
// The 16x1024x1024 GEMM uses V_WMMA_F32_16X16X4_F32 (M=16 fits one tile).
// ---------------------------------------------------------------------------

typedef float v2f __attribute__((ext_vector_type(2)));
typedef float v4f __attribute__((ext_vector_type(4)));
typedef float v8f __attribute__((ext_vector_type(8)));

static constexpr int kBt   = 16;
static constexpr int kN    = 4096;
static constexpr int kFin  = 1024;
static constexpr int kFout = 1024;

static constexpr int kGroups       = 32;            // row-groups per batch (phase 1)
static constexpr int kRowsPerGroup = kN / kGroups;  // 128
static constexpr int kBcastRows    = 64;            // rows per broadcast block

// Workspace layout (floats):
static constexpr size_t kOffPart  = 0;                                  // [16][32][1024]
static constexpr size_t kOffMeans = kOffPart  + (size_t)kBt * kGroups * kFin; // [16][1024]
static constexpr size_t kOffRow   = kOffMeans + (size_t)kBt * kFin;           // [16][1024]

// ---------------------------------------------------------------------------
// Phase 1: partial column sums of X. grid=(kGroups, kBt), block=256.
// Thread t owns columns 4t..4t+3 (float4, coalesced); streams 128 rows with
// non-temporal loads (X is read exactly once -> don't pollute L2).
// ---------------------------------------------------------------------------
__global__ void __launch_bounds__(256)
colsum_partial_kernel(const float* __restrict__ X, float* __restrict__ ws)
{
    const int b = blockIdx.y;
    const int g = blockIdx.x;
    const int t = threadIdx.x;  // 0..255

    const v4f* Xr = reinterpret_cast<const v4f*>(
        X + ((size_t)b * kN + (size_t)g * kRowsPerGroup) * kFin);

    v4f acc = {0.f, 0.f, 0.f, 0.f};
    #pragma unroll 4
    for (int r = 0; r < kRowsPerGroup; ++r) {
        v4f v = __builtin_nontemporal_load(&Xr[(size_t)r * (kFin / 4) + t]);
        acc += v;
    }

    v4f* P = reinterpret_cast<v4f*>(ws + kOffPart + ((size_t)b * kGroups + g) * kFin);
    P[t] = acc;
}

// ---------------------------------------------------------------------------
// Phase 2: reduce the 32 partials per (batch, column) and scale by 1/N.
// grid=kBt, block=256.
// ---------------------------------------------------------------------------
__global__ void __launch_bounds__(256)
reduce_means_kernel(float* __restrict__ ws)
{
    const int b = blockIdx.x;
    const int t = threadIdx.x;

    const v4f* P = reinterpret_cast<const v4f*>(ws + kOffPart + (size_t)b * kGroups * kFin);
    v4f acc = {0.f, 0.f, 0.f, 0.f};
    #pragma unroll
    for (int g = 0; g < kGroups; ++g)
        acc += P[(size_t)g * (kFin / 4) + t];

    const float inv_n = 1.0f / (float)kN;
    v4f* M = reinterpret_cast<v4f*>(ws + kOffMeans + (size_t)b * kFin);
    M[t] = acc * inv_n;
}

// ---------------------------------------------------------------------------
// Phase 3: rowres[16][1024] = means[16][1024] @ B[1024][1024] + cT.
// M=16 is exactly one WMMA tile; one wave (32 lanes) per 16-wide N-tile,
// K marched in steps of 4 via V_WMMA_F32_16X16X4_F32 (full f32 precision).
// grid=64, block=32 (EXEC all-ones as WMMA requires).
//
// A(16x4 f32) layout: lane l -> M = l&15, K pair k0 = (l>>4)*2  (aligned v2f).
// B(4x16 f32) layout: lane l -> N = l&15, rows K = k0, k0+1.
// C/D(16x16 f32):     VGPR v -> row v + (l>=16 ? 8 : 0), col l&15.
// ---------------------------------------------------------------------------
__global__ void __launch_bounds__(32)
wmma_rowgemm_kernel(const float* __restrict__ Bm,
                    const float* __restrict__ cT,
                    float* __restrict__ ws)
{
    const float* __restrict__ means = ws + kOffMeans;
    float* __restrict__ rowres      = ws + kOffRow;

    const int j    = blockIdx.x;        // N-tile: columns 16j .. 16j+15
    const int lane = threadIdx.x;       // 0..31
    const int m    = lane & 15;
    const int k0   = (lane >> 4) * 2;   // 0 or 2
    const int n    = j * 16 + (lane & 15);

    v8f c = {};
    for (int kb = 0; kb < kFin; kb += 4) {
        v2f a = *reinterpret_cast<const v2f*>(means + (size_t)m * kFin + kb + k0);
        v2f bv;
        bv.x = Bm[(size_t)(kb + k0)     * kFout + n];
        bv.y = Bm[(size_t)(kb + k0 + 1) * kFout + n];
        c = __builtin_amdgcn_wmma_f32_16x16x4_f32(
                /*neg_a=*/false, a, /*neg_b=*/false, bv,
                /*c_mod=*/(short)0, c, /*reuse_a=*/false, /*reuse_b=*/false);
    }

    const float bias  = cT[n];
    const int   rbase = (lane >> 4) * 8;
    #pragma unroll
    for (int v = 0; v < 8; ++v)
        rowres[(size_t)(rbase + v) * kFout + n] = c[v] + bias;
}

// ---------------------------------------------------------------------------
// Phase 4: broadcast rowres[b][:] to all 4096 output rows of batch b.
// grid=(kN/kBcastRows, kBt), block=256. Each lane keeps its float4 of the row
// in registers and issues 64 non-temporal coalesced stores (512B per wave).
// ---------------------------------------------------------------------------
__global__ void __launch_bounds__(256)
broadcast_out_kernel(const float* __restrict__ ws, float* __restrict__ out)
{
    const int b  = blockIdx.y;
    const int t  = threadIdx.x;             // 0..255 -> columns 4t..4t+3
    const int n0 = blockIdx.x * kBcastRows;

    const v4f rv = reinterpret_cast<const v4f*>(ws + kOffRow + (size_t)b * kFout)[t];

    v4f* O = reinterpret_cast<v4f*>(out + ((size_t)b * kN + n0) * kFout);
    #pragma unroll 8
    for (int r = 0; r < kBcastRows; ++r)
        __builtin_nontemporal_store(rv, &O[(size_t)r * (kFout / 4) + t]);
}

// ---------------------------------------------------------------------------
extern "C" void kernel_launch(void* const* d_in, const int* in_sizes, int n_in,
                              void* d_out, int out_size, void* d_ws, size_t ws_size,
                              hipStream_t stream)
{
    (void)in_sizes; (void)n_in; (void)out_size; (void)ws_size;

    const float* X  = reinterpret_cast<const float*>(d_in[0]);
    const float* Bm = reinterpret_cast<const float*>(d_in[1]);
    const float* cT = reinterpret_cast<const float*>(d_in[2]);
    float*       out = reinterpret_cast<float*>(d_out);
    float*       ws  = reinterpret_cast<float*>(d_ws);

    // Phase 1: partial column sums (streams all 256MB of X).
    colsum_partial_kernel<<<dim3(kGroups, kBt), 256, 0, stream>>>(X, ws);

    // Phase 2: finish reduction, scale to means.
    reduce_means_kernel<<<dim3(kBt), 256, 0, stream>>>(ws);

    // Phase 3: 16x1024 @ 1024x1024 + cT via f32 WMMA.
    wmma_rowgemm_kernel<<<dim3(kFout / 16), 32, 0, stream>>>(Bm, cT, ws);

    // Phase 4: broadcast the 16 result rows to the full 256MB output.
    broadcast_out_kernel<<<dim3(kN / kBcastRows, kBt), 256, 0, stream>>>(ws, out);
}